// PointToBEV_6614249636322
// MI455X (gfx1250) — compile-verified
//
#include <hip/hip_runtime.h>
#include <hip/hip_bf16.h>

// ---- problem geometry (from reference) ----
#define NPTS   2097152
// crop: bev[1:-1, 101:-101, 101:-101] of (42, 802, 702) -> (40, 600, 500)
#define CZ     40
#define CY     600
#define CX     500
#define NCROP  (CZ * CY * CX)            // 12,000,000 voxels in the crop

#define THREADS        256
#define PTS_PER_BLOCK  1024              // 4 points per thread
#define FLOATS_PER_BLK (PTS_PER_BLOCK * 3)   // 3072 floats = 12 KB LDS tile

typedef unsigned int v4u __attribute__((ext_vector_type(4)));
typedef int          v8i __attribute__((ext_vector_type(8)));
typedef int          v4i __attribute__((ext_vector_type(4)));

// ---------------- kernel 1: scatter (the hot loop) ----------------
// One block = 1024 points = 12 KB, staged into LDS by ONE Tensor Data Mover DMA.
__global__ __launch_bounds__(THREADS) void scatter_kernel(const float* __restrict__ pc,
                                                          float* __restrict__ sums,
                                                          float* __restrict__ cnts) {
    __shared__ float sh[FLOATS_PER_BLK];   // only LDS object -> LDS byte address 0
    const int t = threadIdx.x;
    const float* gbase = pc + (size_t)blockIdx.x * FLOATS_PER_BLK;

#if defined(__gfx1250__)
    // Wave 0 issues a single TDM descriptor: 12KB global -> LDS (1D copy as 2D D#).
    if (t < 32) {
        const unsigned long long ga = (unsigned long long)(uintptr_t)gbase;
        v4u g0;
        g0.x = 1u;                                      // count=1 (valid), user mode
        g0.y = 0u;                                      // lds_addr = 0
        g0.z = (unsigned)(ga & 0xffffffffull);          // global_addr[31:0]
        g0.w = (unsigned)((ga >> 32) & 0x01ffffffull)   // global_addr[56:32]
             | (2u << 30);                              // type = 2 ("image")
        v8i g1;
        g1[0] = (2 << 16);                              // data_size=2 (4B); mask/pad/iter=0
        g1[1] = (FLOATS_PER_BLK & 0xffff) << 16;        // tensor_dim0[15:0]
        g1[2] = (1 << 16);                              // tensor_dim0[31:16]=0 | tensor_dim1=1
        g1[3] = (FLOATS_PER_BLK << 16);                 // tensor_dim1[31:16]=0 | tile_dim0
        g1[4] = 1;                                      // tile_dim1=1 | tile_dim2=0 (unused)
        g1[5] = FLOATS_PER_BLK;                         // tensor_dim0_stride[31:0]
        g1[6] = 0;                                      // stride hi | tensor_dim1_stride lo
        g1[7] = 0;                                      // tensor_dim1_stride hi
        v4i g2 = (v4i)0;                                // dims >=2 unused
        v4i g3 = (v4i)0;
        v8i g4 = (v8i)0;                                // 6-arg builtin: extra group, unused
        __builtin_amdgcn_tensor_load_to_lds(g0, g1, g2, g3, g4, /*cpol=*/0);
        __builtin_amdgcn_s_wait_tensorcnt(0);           // s_wait_tensorcnt 0
    }
#else
    // host-pass / non-gfx1250 typecheck fallback
    for (int k = t; k < FLOATS_PER_BLK / 4; k += THREADS)
        ((float4*)sh)[k] = ((const float4*)gbase)[k];
#endif
    // Escape hatch: the TDM builtin's memory effects don't alias `sh` in LLVM's
    // view, so without this the compiler proves `sh` is never written, folds the
    // ds_loads to undef, and deletes the whole compute loop (observed in round 3).
    asm volatile("" : : "r"(&sh[0]) : "memory");
    __syncthreads();

#pragma unroll
    for (int k = 0; k < PTS_PER_BLOCK / THREADS; ++k) {
        const int p = t + k * THREADS;
        // stride-3 dword reads: gcd(3,64)=1 -> bank-conflict-free across the wave
        const float px = sh[3 * p + 0];
        const float py = sh[3 * p + 1];
        const float pz = sh[3 * p + 2];

        // validity per reference
        if (!(px >= 0.f && px < 70.f && py >= -40.f && py < 40.f && pz >= -3.f && pz < 1.f))
            continue;

        // q = floor(p / 0.1) (+ offsets). Valid points never need the clip.
        const int qx = (int)floorf(px * 10.f) + 1;     // in [1,700]
        const int qy = (int)floorf(py * 10.f) + 401;   // in [1,800]
        const int qz = (int)floorf(pz * 10.f) + 31;    // in [1,40]

        // drop points whose voxel lies outside the cropped output — never read
        if (qx < 101 || qx > 600 || qy < 101 || qy > 700) continue;

        // analytic voxel centers (linspace step is exactly 0.1)
        const float cx = -0.05f  + 0.1f * (float)qx;
        const float cy = -40.05f + 0.1f * (float)qy;
        const float cz = -3.05f  + 0.1f * (float)qz;
        const float dx = cx - px, dy = cy - py, dz = cz - pz;
        const float w  = __expf(-100.f * (dx * dx + dy * dy + dz * dz)); // v_exp_f32

        const int idx = (qz - 1) * (CY * CX) + (qy - 101) * CX + (qx - 101);

        // hardware FP32 atomics (global_atomic_add_f32); 96MB scratch lives in L2
        unsafeAtomicAdd(&sums[idx], w);
        unsafeAtomicAdd(&cnts[idx], 1.0f);
    }
}

// ---------------- kernel 0: zero the scratch (sums + counts) ----------------
__global__ __launch_bounds__(256) void zero_ws_kernel(float4* __restrict__ ws, int n4) {
    int i = blockIdx.x * blockDim.x + threadIdx.x;
    int stride = gridDim.x * blockDim.x;
    for (; i < n4; i += stride) ws[i] = make_float4(0.f, 0.f, 0.f, 0.f);
}

// ---------------- kernel 2: finalize mean = sum / max(count, 1) ----------------
__global__ __launch_bounds__(256) void finalize_kernel(const float4* __restrict__ sums,
                                                       const float4* __restrict__ cnts,
                                                       float4* __restrict__ out, int n4) {
    int i = blockIdx.x * blockDim.x + threadIdx.x;
    if (i >= n4) return;
    float4 s = sums[i];
    float4 c = cnts[i];
    float4 r;
    r.x = s.x / fmaxf(c.x, 1.f);
    r.y = s.y / fmaxf(c.y, 1.f);
    r.z = s.z / fmaxf(c.z, 1.f);
    r.w = s.w / fmaxf(c.w, 1.f);
    out[i] = r;
}

extern "C" void kernel_launch(void* const* d_in, const int* in_sizes, int n_in,
                              void* d_out, int out_size, void* d_ws, size_t ws_size,
                              hipStream_t stream) {
    const float* pc = (const float*)d_in[0];
    // d_in[1] = grid_3D_extended: intentionally unused — voxel centers are affine in the
    // voxel index (linspace step is exactly 0.1), deleting a 283MB random gather.

    float* sums = (float*)d_ws;            // NCROP floats
    float* cnts = sums + NCROP;            // NCROP floats  (total 96 MB scratch, fits L2)
    float* out  = (float*)d_out;           // NCROP floats

    // zero scratch every call (deterministic; harness does not re-poison)
    zero_ws_kernel<<<2048, 256, 0, stream>>>((float4*)d_ws, (2 * NCROP) / 4);

    // 1024 points per block, one TDM DMA per block
    scatter_kernel<<<NPTS / PTS_PER_BLOCK, THREADS, 0, stream>>>(pc, sums, cnts);

    // vectorized finalize (NCROP divisible by 4)
    const int n4 = NCROP / 4;
    finalize_kernel<<<(n4 + 255) / 256, 256, 0, stream>>>(
        (const float4*)sums, (const float4*)cnts, (float4*)out, n4);
}